// FDUnsortedSegmentSum_12661563588635
// MI455X (gfx1250) — compile-verified
//
#include <hip/hip_runtime.h>

// Problem constants (from the reference)
#define NUM_SEGMENTS 10000
#define B_DIM 4
#define S_DIM 200000
#define D_DIM 64

// Native clang vector type (HIP's float4 is a class and is rejected by
// __builtin_nontemporal_load; this one lowers to a single b128 access).
typedef __attribute__((ext_vector_type(4))) float v4f;

// ---------------------------------------------------------------------------
// Kernel 1: zero the output. Harness poisons d_out with 0xAA and we accumulate
// into it, so every kernel_launch call must re-zero. v4f stores ->
// global_store_b128, ~10.24 MB. Regular-temporal stores: we WANT these lines
// resident in L2, because every subsequent atomic RMW targets them.
// ---------------------------------------------------------------------------
__global__ void FDSegSum_zero_kernel(v4f* __restrict__ out, int n4) {
    int t = blockIdx.x * blockDim.x + threadIdx.x;
    if (t < n4) out[t] = (v4f){0.f, 0.f, 0.f, 0.f};
}

// Non-returning device-scope hardware FP32 atomic add.
// WGP$ is read-only on CDNA5, so the RMW resolves at the L2 atomic units.
// Explicit scope:SCOPE_DEV guarantees device visibility; the non-returning
// form is tracked with STOREcnt and drained by S_ENDPGM's implicit wait-idle.
__device__ __forceinline__ void atomic_add_f32_dev(float* addr, float val) {
    asm volatile("global_atomic_add_f32 %0, %1, off scope:SCOPE_DEV"
                 :
                 : "v"(addr), "v"(val)
                 : "memory");
}

// ---------------------------------------------------------------------------
// Kernel 2: scatter-add. One thread per (row i, 16-byte chunk d4).
// 16 threads cover one 64-float row -> v loads are coalesced global_load_b128.
//
// Cache policy is the key optimization:
//   * v (204.8 MB, read exactly once, larger than the 192 MB L2) is loaded
//     NON-TEMPORAL so it claims no L2 capacity.
//   * the output (10.24 MB) therefore stays L2-resident, so all 51.2M atomic
//     RMWs resolve in the L2 atomic units and only ~10 MB writes back to HBM.
//   * segment_index (800 KB, each line read by 16 lanes) is cached normally
//     and prefetched ahead (gfx1250 global_prefetch_b8).
//
// HBM traffic ~= 205 (v) + ~1 (seg) + 10 (zero) + 10 (writeback) MB
//             ~= 226 MB -> ~10 us at 23.3 TB/s.
// ---------------------------------------------------------------------------
__global__ void FDSegSum_scatter_kernel(const float* __restrict__ v,
                                        const int* __restrict__ seg,
                                        float* __restrict__ out) {
    const int t  = blockIdx.x * blockDim.x + threadIdx.x;
    const int i  = t >> 4;           // source row in [0, S)
    const int d4 = (t & 15) << 2;    // float offset within the 64-wide row
    if (i >= S_DIM) return;

    // Pull the (cached, reused) index stream ahead: gfx1250 global_prefetch_b8.
    __builtin_prefetch(seg + i + 4096, 0, 1);

    const int s = seg[i];            // segment id, reused across all batches

    const size_t voff = (size_t)i * D_DIM + d4;
    const size_t ooff = (size_t)s * D_DIM + d4;

#pragma unroll
    for (int b = 0; b < B_DIM; ++b) {
        // Non-temporal 128-bit load: v is streamed once, keep it out of L2.
        const v4f val = __builtin_nontemporal_load(
            (const v4f*)(v + (size_t)b * S_DIM * D_DIM + voff));
        float* o = out + (size_t)b * NUM_SEGMENTS * D_DIM + ooff;
        atomic_add_f32_dev(o + 0, val.x);
        atomic_add_f32_dev(o + 1, val.y);
        atomic_add_f32_dev(o + 2, val.z);
        atomic_add_f32_dev(o + 3, val.w);
    }
}

// ---------------------------------------------------------------------------
// Launch: inputs are (data, v, segment_index); `data` is unused by the math
// (it only feeds a shape assertion in the reference).
// ---------------------------------------------------------------------------
extern "C" void kernel_launch(void* const* d_in, const int* in_sizes, int n_in,
                              void* d_out, int out_size, void* d_ws, size_t ws_size,
                              hipStream_t stream) {
    (void)in_sizes; (void)n_in; (void)d_ws; (void)ws_size;

    const float* v   = (const float*)d_in[1];
    const int*   seg = (const int*)d_in[2];
    float*       out = (float*)d_out;

    // Zero the output: out_size = B * NUM_SEGMENTS * D = 2,560,000 floats.
    const int n4 = out_size / 4;                       // 640,000 v4f
    const int zb = (n4 + 255) / 256;                   // 2,500 blocks
    FDSegSum_zero_kernel<<<zb, 256, 0, stream>>>((v4f*)out, n4);

    // Scatter: S * 16 threads = 3,200,000 -> 12,500 blocks of 256
    // (8 wave32 waves per block).
    const int total  = S_DIM * 16;
    const int blocks = (total + 255) / 256;
    FDSegSum_scatter_kernel<<<blocks, 256, 0, stream>>>(v, seg, out);
}